// VoronoiWatershedLoss_55551107006894
// MI455X (gfx1250) — compile-verified
//
#include <hip/hip_runtime.h>
#include <math.h>

// ---------------------------------------------------------------------------
// VoronoiWatershedLoss for MI455X (gfx1250, wave32, WMMA)
//
// Stage B (dominant) is reformulated as a GEMM:
//   q[p][j] = f(p) . c(j),  f = [x^2, xy, y^2, x, y, 1, 0, 0]
// computed with V_WMMA_F32_16X16X4_F32 (K=4, two per 16x16 tile).
// argmax of exp(-q/2) == argmin of q, so exp is only evaluated once per pixel.
// ---------------------------------------------------------------------------

#define JG       64
#define HH       1024
#define WW       1024
#define HD       512
#define WD       512
#define TOPK_K   61          // ceil(64 * 0.95)

// workspace layout (float units)
#define WS_COEFF 0           // 64*8 floats: quad coefficients, K-padded to 8
#define WS_V     512         // 64*4 floats: eigenvectors column-major
#define WS_LEV   768         // 64*2 floats: eigenvalues ascending
#define WS_MAXS  896         // 64*2 u32:  float-bit max accumulators
#define WS_PRES  1024        // 2 u32:     per-Gaussian presence bitmask
#define WS_VAL   2048        // 512*512 floats
#define WS_VOR   (2048 + HD*WD)  // 512*512 ints

typedef __attribute__((ext_vector_type(2))) float v2f;
typedef __attribute__((ext_vector_type(8))) float v8f;

// ---------------------------------------------------------------------------
// Kernel A: closed-form 2x2 eigendecomposition, scaled inverse covariance,
// quad-form coefficients; zero the reduction accumulators.
// ---------------------------------------------------------------------------
__global__ void vwl_prep_kernel(const float* __restrict__ mu,
                                const float* __restrict__ sigma,
                                float* __restrict__ ws) {
    int t = threadIdx.x;
    unsigned* wsu = (unsigned*)ws;
    if (t < JG * 2) wsu[WS_MAXS + t] = 0u;
    if (t < 2)      wsu[WS_PRES + t] = 0u;
    if (t >= JG) return;

    float a = sigma[t * 4 + 0];
    float b = sigma[t * 4 + 1];
    float c = sigma[t * 4 + 2];
    float d = sigma[t * 4 + 3];
    float bo = 0.5f * (b + c);                 // symmetric by construction

    // eigh ascending: l1 <= l2
    float mid  = 0.5f * (a + d);
    float hdif = 0.5f * (a - d);
    float disc = sqrtf(hdif * hdif + bo * bo);
    float l1 = mid - disc;
    float l2 = mid + disc;

    // eigenvector for l1 (sign irrelevant: used only under abs / quadratics)
    float vx, vy;
    if (fabsf(bo) > 1e-20f) { vx = bo; vy = l1 - a; }
    else if (a <= d)        { vx = 1.0f; vy = 0.0f; }
    else                    { vx = 0.0f; vy = 1.0f; }
    float nrm = rsqrtf(vx * vx + vy * vy);
    vx *= nrm; vy *= nrm;
    float wx = -vy, wy = vx;                   // eigenvector for l2

    ws[WS_V + t * 4 + 0] = vx;                 // V[0][0]
    ws[WS_V + t * 4 + 1] = vy;                 // V[1][0]
    ws[WS_V + t * 4 + 2] = wx;                 // V[0][1]
    ws[WS_V + t * 4 + 3] = wy;                 // V[1][1]
    ws[WS_LEV + t * 2 + 0] = l1;
    ws[WS_LEV + t * 2 + 1] = l2;

    // sg = (4096 / (4*sqrt(det))) * sigma  =>  inv(sg) = sigma^-1 * sqrt(det)/1024
    //   => inv(sg) = [[d,-b],[-b,a]] / (1024*sqrt(det))
    float det = l1 * l2;                       // == a*d - b*c
    float r   = 1.0f / (1024.0f * sqrtf(fmaxf(det, 1e-30f)));
    float Ia = d * r, Ib = -bo * r, Id = a * r;

    float mx = rintf(mu[t * 2 + 0] * 0.5f);    // jnp.round is half-to-even == rintf
    float my = rintf(mu[t * 2 + 1] * 0.5f);

    // q = Ia*(x-mx)^2 + 2 Ib (x-mx)(y-my) + Id*(y-my)^2 expanded in
    // f = [x^2, x*y, y^2, x, y, 1, 0, 0]
    float* cf = ws + WS_COEFF + t * 8;
    cf[0] = Ia;
    cf[1] = 2.0f * Ib;
    cf[2] = Id;
    cf[3] = -2.0f * (Ia * mx + Ib * my);
    cf[4] = -2.0f * (Ib * mx + Id * my);
    cf[5] = Ia * mx * mx + 2.0f * Ib * mx * my + Id * my * my;
    cf[6] = 0.0f;
    cf[7] = 0.0f;
}

// ---------------------------------------------------------------------------
// Kernel B: per wave, 16 pixels x 64 Gaussians via 8x V_WMMA_F32_16X16X4_F32,
// then shfl-butterfly (min,argmin), one exp per pixel.
// A layout (f32 16x4): lanes 0-15 hold K=0(v0),K=1(v1); lanes 16-31 K=2,K=3.
// B layout (f32 4x16): v0 = rows K=0 / K=2 striped over lane halves, v1 = K=1/K=3.
// C layout: VGPR e of lane L -> (M = e + 8*(L>=16), N = L%16).
// ---------------------------------------------------------------------------
__global__ __launch_bounds__(256) void vwl_gauss_argmin_kernel(float* __restrict__ ws) {
    const float* coeff = ws + WS_COEFF;
    float* val = ws + WS_VAL;
    int*   vor = (int*)(ws + WS_VOR);

    int lane = threadIdx.x & 31;
    int wave = blockIdx.x * (blockDim.x >> 5) + (threadIdx.x >> 5);
    int row  = wave >> 5;                      // 512 rows
    int ct   = wave & 31;                      // 32 column tiles of 16
    int c0   = ct * 16;
    int Ml   = lane & 15;
    bool hi  = lane >= 16;

    const float S = 512.0f / 511.0f;           // jnp.linspace(0, 512, 512) step
    float X = (float)(c0 + Ml) * S;
    float Y = (float)row * S;

    v2f a1, a2;
    a1.x = hi ? Y * Y : X * X;                 // K=2 : K=0
    a1.y = hi ? X     : X * Y;                 // K=3 : K=1
    a2.x = hi ? 0.0f  : Y;                     // K=6 : K=4
    a2.y = hi ? 0.0f  : 1.0f;                  // K=7 : K=5

    float bv[8];
    int   bi[8];

#pragma unroll
    for (int t = 0; t < 4; ++t) {
        int j = t * 16 + Ml;                   // this lane's N column
        const float* cf = coeff + j * 8;
        v2f b1, b2;
        b1.x = hi ? cf[2] : cf[0];
        b1.y = hi ? cf[3] : cf[1];
        b2.x = hi ? cf[6] : cf[4];
        b2.y = hi ? cf[7] : cf[5];

        v8f acc = {};
        acc = __builtin_amdgcn_wmma_f32_16x16x4_f32(false, a1, false, b1,
                                                    (short)0, acc, false, false);
        acc = __builtin_amdgcn_wmma_f32_16x16x4_f32(false, a2, false, b2,
                                                    (short)0, acc, false, false);
#pragma unroll
        for (int e = 0; e < 8; ++e) {
            float v = acc[e];
            if (t == 0)        { bv[e] = v; bi[e] = j; }
            else if (v < bv[e]) { bv[e] = v; bi[e] = j; }  // strict <: earlier tile wins ties
        }
    }

    // min+argmin over the 16 lanes of each half (N dimension of the tile)
#pragma unroll
    for (int m = 1; m < 16; m <<= 1) {
#pragma unroll
        for (int e = 0; e < 8; ++e) {
            float ov = __shfl_xor(bv[e], m, 32);
            int   oi = __shfl_xor(bi[e], m, 32);
            if (ov < bv[e] || (ov == bv[e] && oi < bi[e])) { bv[e] = ov; bi[e] = oi; }
        }
    }

    // lane 0 -> pixels M=0..7, lane 16 -> pixels M=8..15
    if (Ml == 0) {
        int base = row * WD + c0 + (hi ? 8 : 0);
#pragma unroll
        for (int e = 0; e < 8; ++e) {
            val[base + e] = __expf(-0.5f * bv[e]);   // max g = exp(-0.5*min q)
            vor[base + e] = bi[e];                   // argmax g = argmin q
        }
    }
}

// ---------------------------------------------------------------------------
// Kernel C: markers at 1024x1024 (bilinear val, laplacian ridges, thresholds),
// per-block LDS float-bit max reduction, then global atomics.
// ---------------------------------------------------------------------------
__global__ __launch_bounds__(256) void vwl_marker_kernel(const float* __restrict__ mu,
                                                         const int* __restrict__ label,
                                                         const float* __restrict__ pos_thres,
                                                         const float* __restrict__ neg_thres,
                                                         float* __restrict__ ws) {
    __shared__ unsigned smax[JG * 2];
    __shared__ unsigned spres[2];
    int t = threadIdx.x;
    for (int i = t; i < JG * 2; i += blockDim.x) smax[i] = 0u;
    if (t < 2) spres[t] = 0u;
    __syncthreads();

    const float* val = ws + WS_VAL;
    const int*   vor = (const int*)(ws + WS_VOR);
    const float* V   = ws + WS_V;

    int idx = blockIdx.x * blockDim.x + t;
    int r = idx >> 10;
    int c = idx & 1023;

    int m0 = vor[(r >> 1) * WD + (c >> 1)];    // 2x2-repeated argmax id, 0..63

    // laplacian (3x3, center -8, zero padding) of the repeated vor field
    int nb = 0;
#pragma unroll
    for (int dr = -1; dr <= 1; ++dr)
#pragma unroll
        for (int dc = -1; dc <= 1; ++dc) {
            if (dr == 0 && dc == 0) continue;
            int rr = r + dr, cc = c + dc;
            int v = 0;
            if (rr >= 0 && rr < HH && cc >= 0 && cc < WW)
                v = vor[(rr >> 1) * WD + (cc >> 1)];
            nb += v;
        }
    bool ridge = (nb - 8 * m0) != 0;

    // bilinear upsample of val: grid = linspace(0, 511, 1024)
    const float SC = 511.0f / 1023.0f;
    float rf = (float)r * SC;
    float cf = (float)c * SC;
    int r0 = (int)rf; if (r0 > HD - 2) r0 = HD - 2;
    int c0 = (int)cf; if (c0 > WD - 2) c0 = WD - 2;
    float fr = rf - (float)r0;
    float fc = cf - (float)c0;
    const float* v0p = val + r0 * WD + c0;
    float x00 = v0p[0], x01 = v0p[1], x10 = v0p[WD], x11 = v0p[WD + 1];
    float top = x00 + (x01 - x00) * fc;
    float bot = x10 + (x11 - x10) * fc;
    float vu  = top + (bot - top) * fr;

    int cls = label[m0];
    int marker;
    if (ridge)                       marker = JG + 1;
    else if (vu < neg_thres[cls])    marker = JG + 1;
    else if (vu < pos_thres[cls])    marker = 0;
    else                             marker = m0 + 1;

    if (marker >= 1 && marker <= JG) {
        int j = marker - 1;
        float rx = (float)c - mu[j * 2 + 0];
        float ry = (float)r - mu[j * 2 + 1];
        const float* Vj = V + j * 4;
        float r0a = fabsf(Vj[0] * rx + Vj[1] * ry);   // |V^T rel| component 0
        float r1a = fabsf(Vj[2] * rx + Vj[3] * ry);   // component 1
        // rot >= 0 => uint bit order == float order
        atomicMax(&smax[2 * j + 0], __float_as_uint(r0a));
        atomicMax(&smax[2 * j + 1], __float_as_uint(r1a));
        atomicOr(&spres[j >> 5], 1u << (j & 31));
    }
    __syncthreads();

    unsigned* gmax  = (unsigned*)ws + WS_MAXS;
    unsigned* gpres = (unsigned*)ws + WS_PRES;
    for (int i = t; i < JG * 2; i += blockDim.x)
        if (smax[i]) atomicMax(&gmax[i], smax[i]);
    if (t < 2 && spres[t]) atomicOr(&gpres[t], spres[t]);
}

// ---------------------------------------------------------------------------
// Kernel D: per-Gaussian loss, mean of 61 smallest via rank counting.
// ---------------------------------------------------------------------------
__global__ void vwl_loss_kernel(const float* __restrict__ ws, float* __restrict__ out) {
    __shared__ float lj[JG];
    __shared__ float ssum;
    int t = threadIdx.x;
    if (t == 0) ssum = 0.0f;
    if (t < JG) {
        const unsigned* gmax  = (const unsigned*)ws + WS_MAXS;
        const unsigned* gpres = (const unsigned*)ws + WS_PRES;
        float l1 = ws[WS_LEV + t * 2 + 0];
        float l2 = ws[WS_LEV + t * 2 + 1];
        bool pres = (gpres[t >> 5] >> (t & 31)) & 1u;
        float m0 = __uint_as_float(gmax[2 * t + 0]);
        float m1 = __uint_as_float(gmax[2 * t + 1]);
        float t0 = pres ? m0 * m0 : l1;        // empty segment -> L_target = Lev
        float t1 = pres ? m1 * m1 : l2;
        float sl0 = sqrtf(fmaxf(l1, 1e-7f)), sl1 = sqrtf(fmaxf(l2, 1e-7f));
        float st0 = sqrtf(fmaxf(t0, 1e-7f)), st1 = sqrtf(fmaxf(t1, 1e-7f));
        float dis = (sl0 - st0) * (sl0 - st0) + (sl1 - st1) * (sl1 - st1);
        lj[t] = log1pf(sqrtf(fmaxf(dis, 1e-7f)));
    }
    __syncthreads();
    if (t < JG) {
        float my = lj[t];
        int rank = 0;
        for (int i = 0; i < JG; ++i) {
            float vi = lj[i];
            rank += (vi < my || (vi == my && i < t)) ? 1 : 0;
        }
        if (rank < TOPK_K) atomicAdd(&ssum, my);
    }
    __syncthreads();
    if (t == 0) out[0] = ssum / (float)TOPK_K;
}

// ---------------------------------------------------------------------------
extern "C" void kernel_launch(void* const* d_in, const int* in_sizes, int n_in,
                              void* d_out, int out_size, void* d_ws, size_t ws_size,
                              hipStream_t stream) {
    (void)in_sizes; (void)n_in; (void)out_size; (void)ws_size;
    const float* mu    = (const float*)d_in[0];
    const float* sigma = (const float*)d_in[1];
    const int*   label = (const int*)d_in[2];
    // d_in[3] (image) is unused by the reference computation.
    const float* pos   = (const float*)d_in[4];
    const float* neg   = (const float*)d_in[5];
    float* ws  = (float*)d_ws;
    float* out = (float*)d_out;

    vwl_prep_kernel<<<1, 128, 0, stream>>>(mu, sigma, ws);
    vwl_gauss_argmin_kernel<<<2048, 256, 0, stream>>>(ws);        // 16384 waves, 16 px each
    vwl_marker_kernel<<<(HH * WW) / 256, 256, 0, stream>>>(mu, label, pos, neg, ws);
    vwl_loss_kernel<<<1, 64, 0, stream>>>(ws, out);
}